// TriangleCell_42726334660721
// MI455X (gfx1250) — compile-verified
//
#include <hip/hip_runtime.h>
#include <hip/hip_bf16.h>

// ============================================================================
// TriangleCell (memory-augmented 4-branch LSTM) for MI455X / gfx1250.
// f32 end-to-end; all GEMMs via V_WMMA_F32_16X16X4_F32 (wave32 WMMA).
// Memory-bound parts (4x 64MB rank-1 memory updates) are single-pass LDS-
// staged fused kernels with b128 loads/stores:
// ~530MB total HBM traffic ~= 23us floor @ 23.3TB/s.
// ============================================================================

#define B_  2048
#define H_  256
#define MD_ 128
#define MK_ 64

typedef __attribute__((ext_vector_type(2))) float v2f;
typedef __attribute__((ext_vector_type(8))) float v8f;

__device__ __forceinline__ float sigmoidf_(float x) { return 1.f / (1.f + __expf(-x)); }

__device__ __forceinline__ float waveReduceSum(float v) {
#pragma unroll
  for (int m = 16; m > 0; m >>= 1) v += __shfl_xor(v, m, 32);
  return v;
}

// ---------------------------------------------------------------------------
// Generic multi-segment GEMM:  C[M,N] = act( sum_s A_s[M,K_s] @ W_s[K_s,N] + b )
// Workgroup tile 128x64, 8 waves (4x2), each wave 2x2 of 16x16 WMMA tiles.
// A staged transposed in LDS (k-major), B staged row-major; K padded w/ zeros.
// K-multiple-of-16 fast path uses unguarded global_load_b128.
// v_wmma_f32_16x16x4_f32 fragment layout (wave32):
//   A 16x4 : lanes 0-15 rows M=0..15, v0=K{0|2}, v1=K{1|3} (hi-half lanes K+2)
//   B 4x16 : lanes 0-15 cols N=0..15, v0=K{0|2}, v1=K{1|3}
//   C/D    : 8 VGPRs, lane<16 -> M=v, lane>=16 -> M=v+8, N=lane&15
// ---------------------------------------------------------------------------
#define GTM 128
#define GTN 64
#define GKB 16

__global__ __launch_bounds__(256) void gemm_f32_wmma(
    const float* __restrict__ A0, int lda0, int K0, const float* __restrict__ W0,
    const float* __restrict__ A1, int lda1, int K1, const float* __restrict__ W1,
    const float* __restrict__ A2, int lda2, int K2, const float* __restrict__ W2,
    const float* __restrict__ bias, const float* __restrict__ bias2,
    float* __restrict__ C, int N, int act)
{
  __shared__ float As[GKB][GTM + 4];
  __shared__ float Bs[GKB][GTN + 4];

  const int tid   = threadIdx.x;
  const int lane  = tid & 31;
  const int wave  = tid >> 5;
  const int waveM = wave >> 1;   // 0..3
  const int waveN = wave & 1;    // 0..1
  const int rowWG = blockIdx.x * GTM;
  const int colWG = blockIdx.y * GTN;
  const int kh    = lane >> 4;   // hi-half of wave
  const int l16   = lane & 15;

  v8f acc[2][2];
#pragma unroll
  for (int i = 0; i < 2; ++i)
#pragma unroll
    for (int j = 0; j < 2; ++j)
#pragma unroll
      for (int v = 0; v < 8; ++v) acc[i][j][v] = 0.f;

  const float* Aseg[3] = {A0, A1, A2};
  const float* Wseg[3] = {W0, W1, W2};
  const int    Ldas[3] = {lda0, lda1, lda2};
  const int    Ks[3]   = {K0, K1, K2};

  for (int s = 0; s < 3; ++s) {
    const int K = Ks[s];
    if (K <= 0) continue;
    const float* A   = Aseg[s];
    const float* W   = Wseg[s];
    const int    lda = Ldas[s];
    for (int k0 = 0; k0 < K; k0 += GKB) {
      const bool full = (k0 + GKB <= K);   // uniform
      { // A tile: 128 rows x 16 k, transposed into LDS (coalesced along K)
        const int m  = tid >> 1;
        const int k8 = (tid & 1) * 8;
        const float* ap = A + (size_t)(rowWG + m) * lda + k0 + k8;
        if (full) {
          const float4 v0 = ((const float4*)ap)[0];   // global_load_b128
          const float4 v1 = ((const float4*)ap)[1];
          As[k8 + 0][m] = v0.x; As[k8 + 1][m] = v0.y;
          As[k8 + 2][m] = v0.z; As[k8 + 3][m] = v0.w;
          As[k8 + 4][m] = v1.x; As[k8 + 5][m] = v1.y;
          As[k8 + 6][m] = v1.z; As[k8 + 7][m] = v1.w;
        } else {                                      // only K=8 GEMMs hit this
#pragma unroll
          for (int j = 0; j < 8; ++j)
            As[k8 + j][m] = (k0 + k8 + j < K) ? ap[j] : 0.f;
        }
      }
      { // B tile: 16 k x 64 n (b128, guard on k-row only)
        const int kk = tid >> 4;
        const int nb = (tid & 15) * 4;
        float4 wv = {0.f, 0.f, 0.f, 0.f};
        if (k0 + kk < K)
          wv = *(const float4*)(W + (size_t)(k0 + kk) * N + colWG + nb);
        Bs[kk][nb + 0] = wv.x; Bs[kk][nb + 1] = wv.y;
        Bs[kk][nb + 2] = wv.z; Bs[kk][nb + 3] = wv.w;
      }
      __syncthreads();
#pragma unroll
      for (int ks = 0; ks < 4; ++ks) {
        const int kfr = ks * 4 + kh * 2;
        v2f afr[2], bfr[2];
#pragma unroll
        for (int sm = 0; sm < 2; ++sm) {
          const int m = waveM * 32 + sm * 16 + l16;
          afr[sm].x = As[kfr + 0][m];
          afr[sm].y = As[kfr + 1][m];
        }
#pragma unroll
        for (int sn = 0; sn < 2; ++sn) {
          const int n = waveN * 32 + sn * 16 + l16;
          bfr[sn].x = Bs[kfr + 0][n];
          bfr[sn].y = Bs[kfr + 1][n];
        }
#pragma unroll
        for (int sm = 0; sm < 2; ++sm)
#pragma unroll
          for (int sn = 0; sn < 2; ++sn)
            acc[sm][sn] = __builtin_amdgcn_wmma_f32_16x16x4_f32(
                false, afr[sm], false, bfr[sn], (short)0, acc[sm][sn], false, false);
      }
      __syncthreads();
    }
  }

  // epilogue: bias(+bias2) + activation (0 none, 1 sigmoid, 2 tanh)
#pragma unroll
  for (int sm = 0; sm < 2; ++sm) {
#pragma unroll
    for (int sn = 0; sn < 2; ++sn) {
      const int col = colWG + waveN * 32 + sn * 16 + l16;
      float bsum = 0.f;
      if (bias)  bsum += bias[col];
      if (bias2) bsum += bias2[col];
#pragma unroll
      for (int v = 0; v < 8; ++v) {
        const int row = rowWG + waveM * 32 + sm * 16 + kh * 8 + v;
        float val = acc[sm][sn][v] + bsum;
        if (act == 1)      val = sigmoidf_(val);
        else if (act == 2) val = tanhf(val);
        C[(size_t)row * N + col] = val;
      }
    }
  }
}

// ---------------------------------------------------------------------------
// LSTM pointwise, float4-vectorized (one thread = 4 consecutive h of one row)
__global__ __launch_bounds__(256) void lstm_pointwise(
    const float* __restrict__ gates, const float* __restrict__ c_old,
    float* __restrict__ h1, float* __restrict__ c_out, float* __restrict__ o_out)
{
  const int i4 = blockIdx.x * blockDim.x + threadIdx.x;   // over B*H/4
  if (i4 >= B_ * H_ / 4) return;
  const int b  = i4 >> 6;           // 64 float4 per row of H=256
  const int h4 = (i4 & 63) * 4;
  const float* g = gates + (size_t)b * 1024;
  const float4 gi = *(const float4*)(g + h4);
  const float4 gf = *(const float4*)(g + 256 + h4);
  const float4 gg = *(const float4*)(g + 512 + h4);
  const float4 go = *(const float4*)(g + 768 + h4);
  const float4 co = ((const float4*)c_old)[i4];
  float4 c, h, o;
  o.x = sigmoidf_(go.x); o.y = sigmoidf_(go.y); o.z = sigmoidf_(go.z); o.w = sigmoidf_(go.w);
  c.x = sigmoidf_(gf.x) * co.x + sigmoidf_(gi.x) * tanhf(gg.x);
  c.y = sigmoidf_(gf.y) * co.y + sigmoidf_(gi.y) * tanhf(gg.y);
  c.z = sigmoidf_(gf.z) * co.z + sigmoidf_(gi.z) * tanhf(gg.z);
  c.w = sigmoidf_(gf.w) * co.w + sigmoidf_(gi.w) * tanhf(gg.w);
  h.x = o.x * c.x; h.y = o.y * c.y; h.z = o.z * c.z; h.w = o.w * c.w;
  ((float4*)c_out)[i4] = c;
  ((float4*)h1)[i4]    = h;
  ((float4*)o_out)[i4] = o;
}

// one wave32 per batch row; replicates torch's raw reshape [B,3,H]->[B,H,3]
__global__ __launch_bounds__(256) void attention_kernel(
    const float* __restrict__ hs1, const float* __restrict__ hm1,
    const float* __restrict__ hl1, const float* __restrict__ attW,
    const float* __restrict__ attb, float* __restrict__ bnew)
{
  const int gw   = (blockIdx.x * blockDim.x + threadIdx.x) >> 5;
  const int lane = threadIdx.x & 31;
  if (gw >= B_) return;
  const float* rows[3] = {hs1 + (size_t)gw * H_, hm1 + (size_t)gw * H_, hl1 + (size_t)gw * H_};
  float logit[3];
#pragma unroll
  for (int r = 0; r < 3; ++r) {
    float p = 0.f;
    for (int h = lane; h < H_; h += 32) p += rows[r][h] * attW[h];
    logit[r] = waveReduceSum(p) + attb[0];
  }
  const float mx  = fmaxf(logit[0], fmaxf(logit[1], logit[2]));
  const float e0  = __expf(logit[0] - mx), e1 = __expf(logit[1] - mx), e2 = __expf(logit[2] - mx);
  const float inv = 1.f / (e0 + e1 + e2);
  const float a[3] = {e0 * inv, e1 * inv, e2 * inv};
  for (int i = lane; i < H_; i += 32) {
    float acc = 0.f;
#pragma unroll
    for (int j = 0; j < 3; ++j) {
      const int u = 3 * i + j;
      const float v = (u < 256) ? rows[0][u] : (u < 512) ? rows[1][u - 256] : rows[2][u - 512];
      acc += v * a[j];
    }
    bnew[(size_t)gw * H_ + i] = acc;
  }
}

// one wave32 per batch row; N=8 is below WMMA tile width -> VALU dot products
__global__ __launch_bounds__(256) void mlp_kernel(
    const float* __restrict__ hw1, const float* __restrict__ bnew,
    const float* __restrict__ event, const float* __restrict__ W1,
    const float* __restrict__ b1, const float* __restrict__ W2,
    const float* __restrict__ b2, float* __restrict__ ebs)
{
  const int gw   = (blockIdx.x * blockDim.x + threadIdx.x) >> 5;
  const int lane = threadIdx.x & 31;
  if (gw >= B_) return;
  float hmid[8];
#pragma unroll
  for (int j = 0; j < 8; ++j) {
    float p = 0.f;
    for (int k = lane; k < 576; k += 32) {
      const float x = (k < 256) ? hw1[(size_t)gw * 256 + k]
                    : (k < 512) ? bnew[(size_t)gw * 256 + (k - 256)]
                                : event[(size_t)gw * 64 + (k - 512)];
      p += x * W1[(size_t)k * 8 + j];
    }
    p = waveReduceSum(p);
    hmid[j] = fmaxf(p + b1[j], 0.f);
  }
  if (lane < 8) {
    float acc = b2[lane];
#pragma unroll
    for (int j = 0; j < 8; ++j) acc += hmid[j] * W2[j * 8 + lane];
    ebs[(size_t)gw * 8 + lane] = acc;
  }
}

// ---------------------------------------------------------------------------
// Fused main-memory: one block per batch row. 32KB LDS tile of m_old is read
// once from HBM (b128); softmax attention read (alpha, r) and rank-1
// erase/add write both come out of LDS.  r_s=r_m=r_l=r_w per the reference.
// ---------------------------------------------------------------------------
__global__ __launch_bounds__(256) void main_mem_kernel(
    const float* __restrict__ m_old, const float* __restrict__ key,
    const float* __restrict__ e, const float* __restrict__ a,
    float* __restrict__ m_new, float* __restrict__ r_w, float* __restrict__ r_s,
    float* __restrict__ r_m, float* __restrict__ r_l)
{
  __shared__ float Mt[MK_ * MD_];   // 32 KB
  __shared__ float kd[MD_], ed[MD_], ad[MD_];
  __shared__ float sc[MK_];
  __shared__ float red[2];
  const int b = blockIdx.x, tid = threadIdx.x;
  const size_t base = (size_t)b * (MK_ * MD_);
  {
    const float4* mb4 = (const float4*)(m_old + base);
    float4* Mt4 = (float4*)Mt;
#pragma unroll
    for (int i = tid; i < MK_ * MD_ / 4; i += 256) Mt4[i] = mb4[i];
  }
  if (tid < MD_) {
    kd[tid] = key[(size_t)b * MD_ + tid];
    ed[tid] = e[(size_t)b * MD_ + tid];
    ad[tid] = a[(size_t)b * MD_ + tid];
  }
  __syncthreads();
  if (tid < MK_) {
    float s = 0.f;
    const float* mr = Mt + tid * MD_;
    for (int d = 0; d < MD_; ++d) s += mr[d] * kd[d];
    sc[tid] = s;
  }
  __syncthreads();
  if (tid == 0) {
    float mx = sc[0];
    for (int k = 1; k < MK_; ++k) mx = fmaxf(mx, sc[k]);
    red[0] = mx;
  }
  __syncthreads();
  if (tid < MK_) sc[tid] = __expf(sc[tid] - red[0]);
  __syncthreads();
  if (tid == 0) {
    float s = 0.f;
    for (int k = 0; k < MK_; ++k) s += sc[k];
    red[1] = 1.f / s;
  }
  __syncthreads();
  if (tid < MK_) sc[tid] *= red[1];            // alpha
  __syncthreads();
  if (tid < MD_) {
    float r = 0.f;
    for (int k = 0; k < MK_; ++k) r += sc[k] * Mt[k * MD_ + tid];
    r_w[(size_t)b * MD_ + tid] = r; r_s[(size_t)b * MD_ + tid] = r;
    r_m[(size_t)b * MD_ + tid] = r; r_l[(size_t)b * MD_ + tid] = r;
  }
  {
    float4* ob4 = (float4*)(m_new + base);
    const float4* Mt4 = (const float4*)Mt;
#pragma unroll
    for (int i = tid; i < MK_ * MD_ / 4; i += 256) {
      const int k  = i >> 5;           // 32 float4 per k-row
      const int d4 = (i & 31) * 4;
      const float al = sc[k];
      const float4 mv = Mt4[i];
      float4 o;
      o.x = mv.x * (1.f - al * ed[d4 + 0]) + al * ad[d4 + 0];
      o.y = mv.y * (1.f - al * ed[d4 + 1]) + al * ad[d4 + 1];
      o.z = mv.z * (1.f - al * ed[d4 + 2]) + al * ad[d4 + 2];
      o.w = mv.w * (1.f - al * ed[d4 + 3]) + al * ad[d4 + 3];
      ob4[i] = o;
    }
  }
}

// ---------------------------------------------------------------------------
// Fused behavior-stress memory: both reads off one bsm_old LDS tile, then a
// single b128 streaming pass over bm_old/sm_old producing bm_new, sm_new, and
// bsm_new = P*bm + (1-P)*sm (384MB through HBM exactly once).
// ---------------------------------------------------------------------------
__global__ __launch_bounds__(256) void bsm_kernel(
    const float* __restrict__ bsm_old, const float* __restrict__ bm_old,
    const float* __restrict__ sm_old,
    const float* __restrict__ key_wbs, const float* __restrict__ key_bbs,
    const float* __restrict__ e_wbs, const float* __restrict__ a_wbs,
    const float* __restrict__ e_bbs, const float* __restrict__ a_bbs,
    float* __restrict__ r_wbs, float* __restrict__ r_bbs,
    float* __restrict__ bm_new, float* __restrict__ sm_new, float* __restrict__ bsm_new)
{
  __shared__ float Mt[MK_ * MD_];   // 32 KB
  __shared__ float kA[MD_], kB[MD_];
  __shared__ float ewb[MD_], awb[MD_], ebb[MD_], abb[MD_];
  __shared__ float alA[MK_], alB[MK_];
  __shared__ float red[4];
  const int b = blockIdx.x, tid = threadIdx.x;
  const size_t base = (size_t)b * (MK_ * MD_);
  {
    const float4* mb4 = (const float4*)(bsm_old + base);
    float4* Mt4 = (float4*)Mt;
#pragma unroll
    for (int i = tid; i < MK_ * MD_ / 4; i += 256) Mt4[i] = mb4[i];
  }
  if (tid < MD_) {
    kA[tid]  = key_wbs[(size_t)b * MD_ + tid];
    kB[tid]  = key_bbs[(size_t)b * MD_ + tid];
    ewb[tid] = e_wbs[(size_t)b * MD_ + tid];
    awb[tid] = a_wbs[(size_t)b * MD_ + tid];
    ebb[tid] = e_bbs[(size_t)b * MD_ + tid];
    abb[tid] = a_bbs[(size_t)b * MD_ + tid];
  }
  __syncthreads();
  if (tid < 128) {
    const int k = tid & 63;
    const float* kk = (tid < 64) ? kA : kB;
    float s = 0.f;
    const float* mr = Mt + k * MD_;
    for (int d = 0; d < MD_; ++d) s += mr[d] * kk[d];
    if (tid < 64) alA[k] = s; else alB[k] = s;
  }
  __syncthreads();
  if (tid < 2) {
    float* al = tid ? alB : alA;
    float mx = al[0];
    for (int k = 1; k < MK_; ++k) mx = fmaxf(mx, al[k]);
    red[tid] = mx;
  }
  __syncthreads();
  if (tid < 128) {
    const int k = tid & 63;
    if (tid < 64) alA[k] = __expf(alA[k] - red[0]);
    else          alB[k] = __expf(alB[k] - red[1]);
  }
  __syncthreads();
  if (tid < 2) {
    float* al = tid ? alB : alA;
    float s = 0.f;
    for (int k = 0; k < MK_; ++k) s += al[k];
    red[2 + tid] = 1.f / s;
  }
  __syncthreads();
  if (tid < 128) {
    const int k = tid & 63;
    if (tid < 64) alA[k] *= red[2]; else alB[k] *= red[3];
  }
  __syncthreads();
  if (tid < MD_) {
    float rA = 0.f, rB = 0.f;
    for (int k = 0; k < MK_; ++k) {
      const float mv = Mt[k * MD_ + tid];
      rA += alA[k] * mv;
      rB += alB[k] * mv;
    }
    r_wbs[(size_t)b * MD_ + tid] = rA;
    r_bbs[(size_t)b * MD_ + tid] = rB;
  }
  __syncthreads();
  {
    const float4* bm4 = (const float4*)(bm_old + base);
    const float4* sm4 = (const float4*)(sm_old + base);
    float4* obm = (float4*)(bm_new + base);
    float4* osm = (float4*)(sm_new + base);
    float4* obs = (float4*)(bsm_new + base);
#pragma unroll
    for (int i = tid; i < MK_ * MD_ / 4; i += 256) {
      const int k  = i >> 5;
      const int d4 = (i & 31) * 4;
      const float aw = alA[k], ab = alB[k];
      const float4 bv = bm4[i];
      const float4 sv = sm4[i];
      float4 bo, so, xo;
      bo.x = bv.x * (1.f - aw * ewb[d4 + 0]) + aw * awb[d4 + 0];
      bo.y = bv.y * (1.f - aw * ewb[d4 + 1]) + aw * awb[d4 + 1];
      bo.z = bv.z * (1.f - aw * ewb[d4 + 2]) + aw * awb[d4 + 2];
      bo.w = bv.w * (1.f - aw * ewb[d4 + 3]) + aw * awb[d4 + 3];
      so.x = sv.x * (1.f - ab * ebb[d4 + 0]) + ab * abb[d4 + 0];
      so.y = sv.y * (1.f - ab * ebb[d4 + 1]) + ab * abb[d4 + 1];
      so.z = sv.z * (1.f - ab * ebb[d4 + 2]) + ab * abb[d4 + 2];
      so.w = sv.w * (1.f - ab * ebb[d4 + 3]) + ab * abb[d4 + 3];
      xo.x = 0.6f * bo.x + 0.4f * so.x;
      xo.y = 0.6f * bo.y + 0.4f * so.y;
      xo.z = 0.6f * bo.z + 0.4f * so.z;
      xo.w = 0.6f * bo.w + 0.4f * so.w;
      obm[i] = bo; osm[i] = so; obs[i] = xo;
    }
  }
}

__global__ __launch_bounds__(256) void final_h_kernel(
    const float* __restrict__ o, const float* __restrict__ c,
    const float* __restrict__ g, const float* __restrict__ u,
    float* __restrict__ hout, int n4)
{
  const int i = blockIdx.x * blockDim.x + threadIdx.x;
  if (i >= n4) return;
  const float4 ov = ((const float4*)o)[i];
  const float4 cv = ((const float4*)c)[i];
  const float4 gv = ((const float4*)g)[i];
  const float4 uv = ((const float4*)u)[i];
  float4 h;
  h.x = ov.x * (cv.x + gv.x * uv.x);
  h.y = ov.y * (cv.y + gv.y * uv.y);
  h.z = ov.z * (cv.z + gv.z * uv.z);
  h.w = ov.w * (cv.w + gv.w * uv.w);
  ((float4*)hout)[i] = h;
}

// ===========================================================================
extern "C" void kernel_launch(void* const* d_in, const int* in_sizes, int n_in,
                              void* d_out, int out_size, void* d_ws, size_t ws_size,
                              hipStream_t stream) {
  (void)in_sizes; (void)n_in; (void)out_size; (void)ws_size;

  // positional tensors (setup_inputs insertion order)
  const float* stress   = (const float*)d_in[0];
  const float* sleep_   = (const float*)d_in[1];
  const float* mood     = (const float*)d_in[2];
  const float* event    = (const float*)d_in[3];
  const float* location = (const float*)d_in[4];
  const float* hw_old   = (const float*)d_in[5];
  const float* cw_old   = (const float*)d_in[6];
  const float* hs_old   = (const float*)d_in[7];
  const float* cs_old   = (const float*)d_in[8];
  const float* hm_old   = (const float*)d_in[9];
  const float* cm_old   = (const float*)d_in[10];
  const float* hl_old   = (const float*)d_in[11];
  const float* cl_old   = (const float*)d_in[12];
  const float* m_old    = (const float*)d_in[13];
  const float* bsm_old  = (const float*)d_in[14];
  const float* bm_old   = (const float*)d_in[15];
  const float* sm_old   = (const float*)d_in[16];
  const float* r_w_old  = (const float*)d_in[17];
  const float* r_s_old  = (const float*)d_in[18];
  const float* r_m_old  = (const float*)d_in[19];
  const float* r_l_old  = (const float*)d_in[20];

  // params dict flattened jax-style (sorted keys), starting at index 21
  const float* att_W    = (const float*)d_in[21];
  const float* att_b    = (const float*)d_in[22];
  const float* h_W      = (const float*)d_in[23];
  const float* h_b      = (const float*)d_in[24];
  const float* hw_W     = (const float*)d_in[25];
  const float* hw_b     = (const float*)d_in[26];
  const float* il_W     = (const float*)d_in[27];
  const float* il_b     = (const float*)d_in[28];
  const float* im_W     = (const float*)d_in[29];
  const float* im_b     = (const float*)d_in[30];
  const float* is_W     = (const float*)d_in[31];
  const float* is_b     = (const float*)d_in[32];
  const float* iw_W     = (const float*)d_in[33];
  const float* iw_b     = (const float*)d_in[34];
  const float* mlp1_W   = (const float*)d_in[35];
  const float* mlp1_b   = (const float*)d_in[36];
  const float* mlp2_W   = (const float*)d_in[37];
  const float* mlp2_b   = (const float*)d_in[38];
  const float* wa_bbs_W = (const float*)d_in[39];
  const float* wa_bbs_b = (const float*)d_in[40];
  const float* wa_w_W   = (const float*)d_in[41];
  const float* wa_w_b   = (const float*)d_in[42];
  const float* wa_wbs_W = (const float*)d_in[43];
  const float* wa_wbs_b = (const float*)d_in[44];
  const float* we_bbs_W = (const float*)d_in[45];
  const float* we_bbs_b = (const float*)d_in[46];
  const float* we_w_W   = (const float*)d_in[47];
  const float* we_w_b   = (const float*)d_in[48];
  const float* we_wbs_W = (const float*)d_in[49];
  const float* we_wbs_b = (const float*)d_in[50];
  const float* wk_bbs_W = (const float*)d_in[51];
  const float* wk_bbs_b = (const float*)d_in[52];
  const float* wk_w_W   = (const float*)d_in[53];
  const float* wk_w_b   = (const float*)d_in[54];
  const float* wk_wbs_W = (const float*)d_in[55];
  const float* wk_wbs_b = (const float*)d_in[56];
  const float* wl_lbs_W = (const float*)d_in[57];
  const float* wl_mbs_W = (const float*)d_in[58];
  const float* wl_sbs_W = (const float*)d_in[59];
  const float* wl_wbs_W = (const float*)d_in[60];
  const float* wp_lbs_W = (const float*)d_in[61];
  const float* wp_mbs_W = (const float*)d_in[62];
  const float* wp_sbs_W = (const float*)d_in[63];
  const float* wp_wbs_W = (const float*)d_in[64];
  const float* wq_lbs_W = (const float*)d_in[65];
  const float* wq_mbs_W = (const float*)d_in[66];
  const float* wq_sbs_W = (const float*)d_in[67];
  const float* wq_wbs_W = (const float*)d_in[68];

  const size_t BH  = (size_t)B_ * H_;          // 524288
  const size_t BKD = (size_t)B_ * MK_ * MD_;   // 16777216
  const size_t BD  = (size_t)B_ * MD_;         // 262144

  // d_out layout: return-order tuple, flat concat
  float* out  = (float*)d_out;
  float* o_hw = out;            float* o_cw = out + BH;
  float* o_hs = out + 2 * BH;   float* o_cs = out + 3 * BH;
  float* o_hm = out + 4 * BH;   float* o_cm = out + 5 * BH;
  float* o_m    = out + 6 * BH;
  float* o_bsm  = o_m + BKD;
  float* o_bm   = o_bsm + BKD;
  float* o_smn  = o_bm + BKD;
  float* o_rw   = o_smn + BKD;
  float* o_rs   = o_rw + BD;
  float* o_rm   = o_rs + BD;
  float* o_hl   = o_rm + BD;
  float* o_cl   = o_hl + BH;
  float* o_rl   = o_cl + BH;

  // workspace (~56 MB of f32)
  float* ws = (float*)d_ws;
  size_t off = 0;
  auto alloc = [&](size_t n) { float* p = ws + off; off += n; return p; };
  float* gates = alloc((size_t)B_ * 1024);
  float* hw1 = alloc(BH); float* hs1 = alloc(BH); float* hm1 = alloc(BH); float* hl1 = alloc(BH);
  float* og_w = alloc(BH); float* og_s = alloc(BH); float* og_m = alloc(BH); float* og_l = alloc(BH);
  float* bnew = alloc(BH);
  float* ebs  = alloc((size_t)B_ * 8);
  float* key_w = alloc(BD);  float* e_w = alloc(BD);  float* a_w = alloc(BD);
  float* key_wbs = alloc(BD); float* e_wbs = alloc(BD); float* a_wbs = alloc(BD);
  float* key_bbs = alloc(BD); float* e_bbs = alloc(BD); float* a_bbs = alloc(BD);
  float* r_wbs = alloc(BD); float* r_bbs = alloc(BD);
  float* g_w = alloc(BH); float* g_s = alloc(BH); float* g_m = alloc(BH); float* g_l = alloc(BH);
  float* u_w = alloc(BH); float* u_s = alloc(BH); float* u_m = alloc(BH); float* u_l = alloc(BH);

  auto gemm = [&](const float* A0, int lda0, int K0, const float* W0,
                  const float* A1, int lda1, int K1, const float* W1,
                  const float* A2, int lda2, int K2, const float* W2,
                  const float* bias, const float* bias2, float* C, int N, int act) {
    dim3 grid(B_ / GTM, N / GTN);
    gemm_f32_wmma<<<grid, dim3(256), 0, stream>>>(A0, lda0, K0, W0, A1, lda1, K1, W1,
                                                  A2, lda2, K2, W2, bias, bias2, C, N, act);
  };

  const int PW4_BLKS  = (int)(BH / 4 / 256);    // float4 pointwise blocks
  const int WAVE_BLKS = B_ * 32 / 256;

  // ---- 1) LSTM branches: gates = [x | r_old | h_old] @ [W_in ; W_h] + b ----
  gemm(stress, 128, 128, iw_W,  r_w_old, 128, 128, iw_W + 128 * 1024,
       hw_old, 256, 256, hw_W,  iw_b, hw_b, gates, 1024, 0);
  lstm_pointwise<<<PW4_BLKS, 256, 0, stream>>>(gates, cw_old, hw1, o_cw, og_w);

  gemm(sleep_, 64, 64, is_W,    r_s_old, 128, 128, is_W + 64 * 1024,
       hs_old, 256, 256, h_W,   is_b, h_b, gates, 1024, 0);
  lstm_pointwise<<<PW4_BLKS, 256, 0, stream>>>(gates, cs_old, hs1, o_cs, og_s);

  gemm(mood, 64, 64, im_W,      r_m_old, 128, 128, im_W + 64 * 1024,
       hm_old, 256, 256, h_W,   im_b, h_b, gates, 1024, 0);
  lstm_pointwise<<<PW4_BLKS, 256, 0, stream>>>(gates, cm_old, hm1, o_cm, og_m);

  gemm(location, 64, 64, il_W,  r_l_old, 128, 128, il_W + 64 * 1024,
       hl_old, 256, 256, h_W,   il_b, h_b, gates, 1024, 0);
  lstm_pointwise<<<PW4_BLKS, 256, 0, stream>>>(gates, cl_old, hl1, o_cl, og_l);

  // ---- 2) attention + MLP ----
  attention_kernel<<<WAVE_BLKS, 256, 0, stream>>>(hs1, hm1, hl1, att_W, att_b, bnew);
  mlp_kernel<<<WAVE_BLKS, 256, 0, stream>>>(hw1, bnew, event, mlp1_W, mlp1_b, mlp2_W, mlp2_b, ebs);

  // ---- 3) key/erase/add projections (WMMA GEMMs with fused activation) ----
  gemm(event, 64, 64, wk_w_W, 0, 0, 0, 0, 0, 0, 0, 0, wk_w_b, nullptr, key_w, 128, 2);
  gemm(ebs,    8,  8, we_w_W, 0, 0, 0, 0, 0, 0, 0, 0, we_w_b, nullptr, e_w,   128, 1);
  gemm(ebs,    8,  8, wa_w_W, 0, 0, 0, 0, 0, 0, 0, 0, wa_w_b, nullptr, a_w,   128, 2);
  gemm(hw1, 256, 256, wk_wbs_W, 0, 0, 0, 0, 0, 0, 0, 0, wk_wbs_b, nullptr, key_wbs, 128, 2);
  gemm(hw1, 256, 256, we_wbs_W, 0, 0, 0, 0, 0, 0, 0, 0, we_wbs_b, nullptr, e_wbs,   128, 1);
  gemm(hw1, 256, 256, wa_wbs_W, 0, 0, 0, 0, 0, 0, 0, 0, wa_wbs_b, nullptr, a_wbs,   128, 2);
  gemm(bnew, 256, 256, wk_bbs_W, 0, 0, 0, 0, 0, 0, 0, 0, wk_bbs_b, nullptr, key_bbs, 128, 2);
  gemm(bnew, 256, 256, we_bbs_W, 0, 0, 0, 0, 0, 0, 0, 0, we_bbs_b, nullptr, e_bbs,   128, 1);
  gemm(bnew, 256, 256, wa_bbs_W, 0, 0, 0, 0, 0, 0, 0, 0, wa_bbs_b, nullptr, a_bbs,   128, 2);

  // ---- 4) fused memory read/writes (single HBM pass per matrix) ----
  main_mem_kernel<<<B_, 256, 0, stream>>>(m_old, key_w, e_w, a_w, o_m, o_rw, o_rs, o_rm, o_rl);
  bsm_kernel<<<B_, 256, 0, stream>>>(bsm_old, bm_old, sm_old, key_wbs, key_bbs,
                                     e_wbs, a_wbs, e_bbs, a_bbs,
                                     r_wbs, r_bbs, o_bm, o_smn, o_bsm);

  // ---- 5) gating: g = sigmoid(c_new@wp + r@wq), u = r@wl ----
  gemm(o_cw, 256, 256, wp_wbs_W, r_wbs, 128, 128, wq_wbs_W, 0, 0, 0, 0, nullptr, nullptr, g_w, 256, 1);
  gemm(r_wbs, 128, 128, wl_wbs_W, 0, 0, 0, 0, 0, 0, 0, 0, nullptr, nullptr, u_w, 256, 0);
  gemm(o_cs, 256, 256, wp_sbs_W, r_bbs, 128, 128, wq_sbs_W, 0, 0, 0, 0, nullptr, nullptr, g_s, 256, 1);
  gemm(r_bbs, 128, 128, wl_sbs_W, 0, 0, 0, 0, 0, 0, 0, 0, nullptr, nullptr, u_s, 256, 0);
  gemm(o_cm, 256, 256, wp_mbs_W, r_bbs, 128, 128, wq_mbs_W, 0, 0, 0, 0, nullptr, nullptr, g_m, 256, 1);
  gemm(r_bbs, 128, 128, wl_mbs_W, 0, 0, 0, 0, 0, 0, 0, 0, nullptr, nullptr, u_m, 256, 0);
  gemm(o_cl, 256, 256, wp_lbs_W, r_bbs, 128, 128, wq_lbs_W, 0, 0, 0, 0, nullptr, nullptr, g_l, 256, 1);
  gemm(r_bbs, 128, 128, wl_lbs_W, 0, 0, 0, 0, 0, 0, 0, 0, nullptr, nullptr, u_l, 256, 0);

  // ---- 6) final hidden states ----
  final_h_kernel<<<PW4_BLKS, 256, 0, stream>>>(og_w, o_cw, g_w, u_w, o_hw, (int)(BH / 4));
  final_h_kernel<<<PW4_BLKS, 256, 0, stream>>>(og_s, o_cs, g_s, u_s, o_hs, (int)(BH / 4));
  final_h_kernel<<<PW4_BLKS, 256, 0, stream>>>(og_m, o_cm, g_m, u_m, o_hm, (int)(BH / 4));
  final_h_kernel<<<PW4_BLKS, 256, 0, stream>>>(og_l, o_cl, g_l, u_l, o_hl, (int)(BH / 4));
}